// GINConvNet_23639499997382
// MI455X (gfx1250) — compile-verified
//
#include <hip/hip_runtime.h>

typedef __attribute__((ext_vector_type(2))) float v2f;
typedef __attribute__((ext_vector_type(8))) float v8f;

#define N_NODES  50000
#define N_EDGES  1600000
#define DIM      128
#define N_GRAPHS 1024
#define BN_EPS   1e-5f

// ---------------- elementwise helpers ----------------
__global__ void k_copy4(const float* __restrict__ src, float* __restrict__ dst, int n4) {
    int i = blockIdx.x * blockDim.x + threadIdx.x;
    if (i < n4) ((float4*)dst)[i] = ((const float4*)src)[i];
}

__global__ void k_zero(float* __restrict__ p, int n) {
    int i = blockIdx.x * blockDim.x + threadIdx.x;
    if (i < n) p[i] = 0.0f;
}

// ---------------- GIN aggregation: agg[dst] += h[src] ----------------
__global__ void k_scatter(const float* __restrict__ h, float* __restrict__ agg,
                          const int* __restrict__ ei, int nE) {
    int idx  = blockIdx.x * blockDim.x + threadIdx.x;
    int e    = idx >> 5;
    int lane = idx & 31;
    if (e >= nE) return;
    int s = ei[e];         // edge_index[0][e]
    int d = ei[nE + e];    // edge_index[1][e]
    const float4 v = *(const float4*)(h + (size_t)s * DIM + lane * 4);
    float* p = agg + (size_t)d * DIM + lane * 4;
    atomicAdd(p + 0, v.x);
    atomicAdd(p + 1, v.y);
    atomicAdd(p + 2, v.z);
    atomicAdd(p + 3, v.w);
}

// ---------------- fp32 WMMA GEMM: C = relu(A[M,128] @ W[128,128] + bias) ----------------
// 256-thread block = 8 waves = 4 M-tiles x 2 N-halves; each wave computes a 16x64 slab
// (4 v8f accumulators) -> ~150 live VGPRs: fits 4 waves/SIMD with no spills, no MSB regs.
// W staged in LDS transposed as XOR-swizzled v2f pairs:
//   WT[n][ (k>>1) ^ ((n&15)<<1) ] = ( W[k][n], W[k+1][n] )   (k even)
// so each b-fragment is a single conflict-free ds_load_b64 in WMMA layout.
__global__ __launch_bounds__(256)
__attribute__((amdgpu_waves_per_eu(4)))
void k_gemm_bias_relu(
    const float* __restrict__ A, const float* __restrict__ W,
    const float* __restrict__ bias, float* __restrict__ C, int M) {
    __shared__ v2f WT[DIM * 64];      // exactly 64 KB

    int tid = threadIdx.x;
    {
        float* wsf = (float*)WT;
        for (int idx = tid; idx < DIM * DIM; idx += 256) {
            int k = idx >> 7;          // row of W (K index), coalesced global read
            int n = idx & 127;         // col of W (N index)
            int slot = (k >> 1) ^ ((n & 15) << 1);
            wsf[n * 128 + slot * 2 + (k & 1)] = W[idx];
        }
    }
    __syncthreads();

    int wave   = tid >> 5;
    int lane   = tid & 31;
    int mtile  = blockIdx.x * 4 + (wave >> 1);
    int nhalf  = wave & 1;             // which 64-column half of N
    int mtiles = M >> 4;               // M == 50000 -> 3125 exact tiles
    if (mtile >= mtiles) return;

    int m0 = mtile << 4;
    int lo = lane & 15;
    int hi = lane >> 4;                // 0 for lanes 0-15, 1 for lanes 16-31
    int cx = hi ^ (lo << 1);           // per-lane swizzle constant

    v8f acc[4];
#pragma unroll
    for (int n = 0; n < 4; ++n) acc[n] = (v8f){0.f,0.f,0.f,0.f,0.f,0.f,0.f,0.f};

    // A fragment: lane holds A[m0+lo][kb], A[m0+lo][kb+1], kb = 4*kc + 2*hi
    const float* arow = A + (size_t)(m0 + lo) * DIM + 2 * hi;
    // B base for this wave's N-half: n16 = (nhalf*4 + n)*16 + lo
    const v2f* wthalf = WT + (size_t)nhalf * 4096 + lo * 64;

    // Two-stage pipeline: preload 16 A fragments, compute 16 kc steps, repeat.
#pragma unroll
    for (int half = 0; half < 2; ++half) {
        v2f afrag[16];
#pragma unroll
        for (int i = 0; i < 16; ++i)
            afrag[i] = *(const v2f*)(arow + 4 * (half * 16 + i));

#pragma unroll
        for (int i = 0; i < 16; ++i) {
            int kc = half * 16 + i;
            const v2f* bp = wthalf + ((kc << 1) ^ cx);   // slot = (2kc+hi)^(lo<<1)
#pragma unroll
            for (int n = 0; n < 4; ++n) {
                acc[n] = __builtin_amdgcn_wmma_f32_16x16x4_f32(
                    false, afrag[i], false, bp[n * 1024], (short)0, acc[n], false, false);
            }
        }
    }

    // C/D layout: VGPR r, lanes 0-15 -> M=m0+r, lanes 16-31 -> M=m0+8+r; N = 64*nhalf+n*16+lo
#pragma unroll
    for (int n = 0; n < 4; ++n) {
        int ncol = nhalf * 64 + n * 16 + lo;
        float bv = bias[ncol];
#pragma unroll
        for (int r = 0; r < 8; ++r) {
            float v = acc[n][r] + bv;
            v = v > 0.0f ? v : 0.0f;
            C[(size_t)(m0 + r + 8 * hi) * DIM + ncol] = v;
        }
    }
}

// ---------------- batchnorm statistics: per-column sum / sumsq ----------------
__global__ void k_colstats(const float* __restrict__ h, float* __restrict__ stats, int rows) {
    int col = threadIdx.x;                 // 128 threads
    int r0  = blockIdx.x * 128;
    int re  = r0 + 128; if (re > rows) re = rows;
    float s = 0.0f, s2 = 0.0f;
    for (int r = r0; r < re; ++r) {
        float v = h[(size_t)r * DIM + col];
        s += v; s2 += v * v;
    }
    atomicAdd(&stats[col], s);
    atomicAdd(&stats[DIM + col], s2);
}

__global__ void k_bn_apply(float* __restrict__ h, const float* __restrict__ stats,
                           const float* __restrict__ gamma, const float* __restrict__ beta,
                           int n) {
    int i = blockIdx.x * blockDim.x + threadIdx.x;
    if (i >= n) return;
    int col = i & (DIM - 1);
    const float inv_n = 1.0f / (float)N_NODES;
    float mu  = stats[col] * inv_n;
    float var = stats[DIM + col] * inv_n - mu * mu;
    float sc  = rsqrtf(var + BN_EPS) * gamma[col];
    h[i] = (h[i] - mu) * sc + beta[col];
}

// ---------------- graph pooling: pooled[batch[node]] += h[node] ----------------
__global__ void k_pool(const float* __restrict__ h, const int* __restrict__ batch,
                       float* __restrict__ pooled, int nNodes) {
    int idx  = blockIdx.x * blockDim.x + threadIdx.x;
    int node = idx >> 5;
    int lane = idx & 31;
    if (node >= nNodes) return;
    int g = batch[node];
    const float4 v = *(const float4*)(h + (size_t)node * DIM + lane * 4);
    float* p = pooled + (size_t)g * DIM + lane * 4;
    atomicAdd(p + 0, v.x);
    atomicAdd(p + 1, v.y);
    atomicAdd(p + 2, v.z);
    atomicAdd(p + 3, v.w);
}

// ---------------- final FC: out[g] = dot(pooled[g], fc_w) + fc_b ----------------
__global__ void k_fc(const float* __restrict__ pooled, const float* __restrict__ w,
                     const float* __restrict__ b, float* __restrict__ out) {
    __shared__ float red[128];
    int g = blockIdx.x, t = threadIdx.x;
    red[t] = pooled[(size_t)g * DIM + t] * w[t];
    __syncthreads();
    for (int s = 64; s > 0; s >>= 1) {
        if (t < s) red[t] += red[t + s];
        __syncthreads();
    }
    if (t == 0) out[g] = red[0] + b[0];
}

// ---------------- driver ----------------
extern "C" void kernel_launch(void* const* d_in, const int* in_sizes, int n_in,
                              void* d_out, int out_size, void* d_ws, size_t ws_size,
                              hipStream_t stream) {
    (void)in_sizes; (void)n_in; (void)out_size; (void)ws_size;

    const float* x     = (const float*)d_in[0];
    const int*   ei    = (const int*)d_in[1];   // [2, N_EDGES] int32
    const int*   batch = (const int*)d_in[2];   // [N_NODES] int32
    // params in dict order: per layer {w1,b1,w2,b2,gamma,beta}, then fc_w, fc_b
    const float* p[24];
    for (int i = 0; i < 24; ++i) p[i] = (const float*)d_in[3 + i];
    const float* fc_w = (const float*)d_in[27];
    const float* fc_b = (const float*)d_in[28];
    float* out = (float*)d_out;

    char* ws = (char*)d_ws;
    float* agg    = (float*)(ws);
    float* bufA   = (float*)(ws + 25600000);
    float* bufB   = (float*)(ws + 51200000);
    float* stats  = (float*)(ws + 76800000);            // 256 floats
    float* pooled = (float*)(ws + 76800000 + 1024);     // 1024*128 floats

    const int NF = N_NODES * DIM;                       // 6,400,000
    const int gemm_blocks = (3125 + 3) / 4;             // 782 (4 M-tiles per block)

    const float* h = x;
    for (int L = 0; L < 4; ++L) {
        const float* w1 = p[L * 6 + 0]; const float* b1 = p[L * 6 + 1];
        const float* w2 = p[L * 6 + 2]; const float* b2 = p[L * 6 + 3];
        const float* ga = p[L * 6 + 4]; const float* be = p[L * 6 + 5];

        k_copy4 <<<(NF / 4 + 255) / 256, 256, 0, stream>>>(h, agg, NF / 4);
        k_scatter<<<(N_EDGES * 32) / 256, 256, 0, stream>>>(h, agg, ei, N_EDGES);
        k_gemm_bias_relu<<<gemm_blocks, 256, 0, stream>>>(agg,  w1, b1, bufA, N_NODES);
        k_gemm_bias_relu<<<gemm_blocks, 256, 0, stream>>>(bufA, w2, b2, bufB, N_NODES);
        k_zero<<<1, 256, 0, stream>>>(stats, 256);
        k_colstats<<<(N_NODES + 127) / 128, 128, 0, stream>>>(bufB, stats, N_NODES);
        k_bn_apply<<<(NF + 255) / 256, 256, 0, stream>>>(bufB, stats, ga, be, NF);
        h = bufB;
    }

    k_zero<<<(N_GRAPHS * DIM + 255) / 256, 256, 0, stream>>>(pooled, N_GRAPHS * DIM);
    k_pool<<<(N_NODES * 32 + 255) / 256, 256, 0, stream>>>(h, batch, pooled, N_NODES);
    k_fc<<<N_GRAPHS, 128, 0, stream>>>(pooled, fc_w, fc_b, out);
}